// LayerGather_76338748719193
// MI455X (gfx1250) — compile-verified
//
#include <hip/hip_runtime.h>

// ---------------- problem constants ----------------
constexpr int HIDDEN = 2048;
constexpr int INTER  = 1408;
constexpr int TOPK   = 6;

constexpr int NSEG1 = 4;              // k-split of HIDDEN for gate_up GEMV (512 each)
constexpr int SEG1  = HIDDEN / NSEG1; // 512
constexpr int NSEG2 = 2;              // k-split of INTER for down GEMV (22 chunks each)
constexpr int CHUNK = 32;             // k-chunk staged in LDS per step
constexpr int PAD   = 36;             // LDS row stride (floats): %4==0 for b128, conflict-free A reads

// workspace layout (floats)
constexpr int P1_OFF    = 0;                          // [6][NSEG1][2*INTER] gate@[r], up@[INTER+r]
constexpr int P1_SIZE   = TOPK * NSEG1 * 2 * INTER;   // 67584
constexpr int INTER_OFF = P1_OFF + P1_SIZE;           // [6][INTER]
constexpr int INTER_SZ  = TOPK * INTER;               // 8448
constexpr int P2_OFF    = INTER_OFF + INTER_SZ;       // [6][NSEG2][HIDDEN]

typedef float v2f __attribute__((ext_vector_type(2)));
typedef float v8f __attribute__((ext_vector_type(8)));

// topk_idx declared int64 in the reference, but JAX w/o x64 delivers int32.
// For int64 little-endian values in [0,60) the odd 32-bit words are all zero.
__device__ __forceinline__ int load_expert(const int* __restrict__ p, int k) {
    const bool is64 = ((p[1] | p[3] | p[5]) == 0);
    int v = is64 ? p[2 * k] : p[k];
    v = v < 0 ? 0 : (v > 59 ? 59 : v);
    return v;
}

// ============ kernel 1: gate_up partial GEMV via V_WMMA_F32_16X16X4_F32 ============
// block = 256 thr (8 waves). Each block: 128 inter-rows (gate+up rows), one k-segment.
// grid = 6 experts * 11 row-tiles * NSEG1 segments = 264 blocks.
__global__ __launch_bounds__(256) void k_gate_up(const float* __restrict__ x,
                                                 const int* __restrict__ topk_idx,
                                                 const float* __restrict__ gate_up_all,
                                                 float* __restrict__ ws) {
    __shared__ float tile[256 * PAD];  // rows 0..127 gate, 128..255 up  (36 KB)
    __shared__ float xs[HIDDEN];       // 8 KB

    const int b       = blockIdx.x;
    const int e       = b / (11 * NSEG1);
    const int rem     = b % (11 * NSEG1);
    const int tileIdx = rem / NSEG1;
    const int s       = rem % NSEG1;
    const int rowBase = tileIdx * 128;

    const int tid  = threadIdx.x;
    const int lane = tid & 31;
    const int wave = tid >> 5;
    const int lm   = lane & 15;
    const int hs   = lane >> 4;                 // 0: K=0,1  1: K=2,3 half of the wave
    const float bm = (lm == 0) ? 1.0f : 0.0f;   // B column-0 mask (lanes 0 and 16 only)

    const int eIdx = load_expert(topk_idx, e);
    const float* __restrict__ gu = gate_up_all + (size_t)eIdx * (2 * INTER) * HIDDEN;

    for (int j = tid; j < HIDDEN / 4; j += 256)
        ((float4*)xs)[j] = ((const float4*)x)[j];

    v8f accg = {0, 0, 0, 0, 0, 0, 0, 0};
    v8f accu = {0, 0, 0, 0, 0, 0, 0, 0};

    const int gRow = wave * 16 + lm;   // this lane's A-matrix row in LDS (gate)
    const int uRow = 128 + gRow;       // (up)

    for (int c = 0; c < SEG1 / CHUNK; ++c) {   // 16 chunks of 32
        const int kbase = s * SEG1 + c * CHUNK;
        __syncthreads();
        // cooperative, coalesced: 256 rows x 32 floats = 2048 float4 / 256 threads
#pragma unroll
        for (int j = 0; j < 8; ++j) {
            const int idx = tid + 256 * j;
            const int r   = idx >> 3;
            const int q   = idx & 7;
            const int guRow = (r < 128) ? (rowBase + r) : (INTER + rowBase + (r - 128));
            const float4 v = *(const float4*)(gu + (size_t)guRow * HIDDEN + kbase + q * 4);
            *(float4*)(tile + r * PAD + q * 4) = v;
        }
        __syncthreads();
#pragma unroll
        for (int kk = 0; kk < CHUNK / 4; ++kk) {
            const int col    = kk * 4 + hs * 2;
            const float2 ag2 = *(const float2*)(tile + gRow * PAD + col);
            const float2 au2 = *(const float2*)(tile + uRow * PAD + col);
            const float2 xv  = *(const float2*)(xs + kbase + kk * 4 + hs * 2);
            v2f a, au, bb;
            a[0]  = ag2.x;    a[1]  = ag2.y;
            au[0] = au2.x;    au[1] = au2.y;
            bb[0] = xv.x * bm; bb[1] = xv.y * bm;
            accg = __builtin_amdgcn_wmma_f32_16x16x4_f32(false, a,  false, bb, (short)0, accg, false, false);
            accu = __builtin_amdgcn_wmma_f32_16x16x4_f32(false, au, false, bb, (short)0, accu, false, false);
        }
    }

    // D column 0: lane 0 holds rows 0..7 in acc[0..7], lane 16 holds rows 8..15
    if (lm == 0) {
        float* P = ws + P1_OFF + (size_t)(e * NSEG1 + s) * (2 * INTER);
        const int r0 = rowBase + wave * 16 + hs * 8;
#pragma unroll
        for (int v = 0; v < 8; ++v) {
            P[r0 + v]         = accg[v];
            P[INTER + r0 + v] = accu[v];
        }
    }
}

// ============ kernel 1b: combine k-split partials, silu(gate)*up ============
__global__ __launch_bounds__(256) void k_silu(float* __restrict__ ws) {
    const int t = blockIdx.x * 256 + threadIdx.x;
    if (t >= TOPK * INTER) return;
    const int e = t / INTER;
    const int i = t % INTER;
    float g = 0.f, u = 0.f;
#pragma unroll
    for (int s = 0; s < NSEG1; ++s) {
        const float* Ps = ws + P1_OFF + (size_t)(e * NSEG1 + s) * (2 * INTER);
        g += Ps[i];
        u += Ps[INTER + i];
    }
    const float sig = 1.0f / (1.0f + expf(-g));
    ws[INTER_OFF + t] = g * sig * u;
}

// ============ kernel 2: down partial GEMV via WMMA ============
// grid = 6 experts * 16 h-tiles * NSEG2 segments = 192 blocks.
__global__ __launch_bounds__(256) void k_down(const int* __restrict__ topk_idx,
                                              const float* __restrict__ down_all,
                                              float* __restrict__ ws) {
    __shared__ float tile[128 * PAD]; // 18 KB
    __shared__ float xs[INTER];       // 5.5 KB: inter vector of this expert

    const int b       = blockIdx.x;
    const int e       = b / (16 * NSEG2);
    const int rem     = b % (16 * NSEG2);
    const int tileIdx = rem / NSEG2;
    const int s       = rem % NSEG2;
    const int rowBase = tileIdx * 128;

    const int tid  = threadIdx.x;
    const int lane = tid & 31;
    const int wave = tid >> 5;
    const int lm   = lane & 15;
    const int hs   = lane >> 4;
    const float bm = (lm == 0) ? 1.0f : 0.0f;

    const int eIdx = load_expert(topk_idx, e);
    const float* __restrict__ dn = down_all + (size_t)eIdx * HIDDEN * INTER;

    for (int j = tid; j < INTER / 4; j += 256)
        ((float4*)xs)[j] = ((const float4*)(ws + INTER_OFF + (size_t)e * INTER))[j];

    v8f acc = {0, 0, 0, 0, 0, 0, 0, 0};
    const int rowLds = wave * 16 + lm;

    const int cBeg = s * 22, cEnd = cBeg + 22;     // 44 chunks total over K=1408
    for (int c = cBeg; c < cEnd; ++c) {
        const int kbase = c * CHUNK;
        __syncthreads();
#pragma unroll
        for (int j = 0; j < 4; ++j) {              // 128 rows x 32 floats
            const int idx = tid + 256 * j;
            const int r   = idx >> 3;
            const int q   = idx & 7;
            const float4 v = *(const float4*)(dn + (size_t)(rowBase + r) * INTER + kbase + q * 4);
            *(float4*)(tile + r * PAD + q * 4) = v;
        }
        __syncthreads();
#pragma unroll
        for (int kk = 0; kk < CHUNK / 4; ++kk) {
            const float2 a2 = *(const float2*)(tile + rowLds * PAD + kk * 4 + hs * 2);
            const float2 xv = *(const float2*)(xs + kbase + kk * 4 + hs * 2);
            v2f a, bb;
            a[0]  = a2.x;      a[1]  = a2.y;
            bb[0] = xv.x * bm; bb[1] = xv.y * bm;
            acc = __builtin_amdgcn_wmma_f32_16x16x4_f32(false, a, false, bb, (short)0, acc, false, false);
        }
    }

    if (lm == 0) {
        float* P = ws + P2_OFF + (size_t)(e * NSEG2 + s) * HIDDEN;
        const int r0 = rowBase + wave * 16 + hs * 8;
#pragma unroll
        for (int v = 0; v < 8; ++v) P[r0 + v] = acc[v];
    }
}

// ============ kernel 3: weighted deterministic reduce over experts/segments ============
__global__ __launch_bounds__(256) void k_reduce(const float* __restrict__ topk_w,
                                                const float* __restrict__ ws,
                                                float* __restrict__ out) {
    const int h = blockIdx.x * 256 + threadIdx.x;
    if (h >= HIDDEN) return;
    float acc = 0.f;
#pragma unroll
    for (int e = 0; e < TOPK; ++e) {
        float v = 0.f;
#pragma unroll
        for (int s = 0; s < NSEG2; ++s)
            v += ws[P2_OFF + (size_t)(e * NSEG2 + s) * HIDDEN + h];
        acc += topk_w[e] * v;
    }
    out[h] = acc;
}

extern "C" void kernel_launch(void* const* d_in, const int* in_sizes, int n_in,
                              void* d_out, int out_size, void* d_ws, size_t ws_size,
                              hipStream_t stream) {
    (void)in_sizes; (void)n_in; (void)out_size; (void)ws_size;
    const float* x        = (const float*)d_in[0];
    const int*   topk_idx = (const int*)d_in[1];
    const float* topk_w   = (const float*)d_in[2];
    const float* gu       = (const float*)d_in[3];
    const float* dn       = (const float*)d_in[4];
    float*       out      = (float*)d_out;
    float*       ws       = (float*)d_ws;

    k_gate_up<<<TOPK * 11 * NSEG1, 256, 0, stream>>>(x, topk_idx, gu, ws);
    k_silu<<<(TOPK * INTER + 255) / 256, 256, 0, stream>>>(ws);
    k_down<<<TOPK * 16 * NSEG2, 256, 0, stream>>>(topk_idx, dn, ws);
    k_reduce<<<(HIDDEN + 255) / 256, 256, 0, stream>>>(topk_w, ws, out);
}